// conv2d_sig_5231270167233
// MI455X (gfx1250) — compile-verified
//
#include <hip/hip_runtime.h>

typedef __attribute__((ext_vector_type(2))) float v2f;
typedef __attribute__((ext_vector_type(8))) float v8f;

#define H_ 20
#define W_ 20
#define HW 400
#define NV 4
#define NK 12
#define NF 17        // NK + NVARS + 1
#define NT 96
#define NB 64
#define SIGLEN 306   // NF + NF*NF
#define NPIX (NB*HW) // 25600

// ---------------------------------------------------------------------------
// Kernel 1: fused conv3x3 + feature/path build (in LDS) + path signature.
// One wave (32 lanes) per pixel (b,h,w); 8 waves (8 pixels) per block.
// lvl2 16x16 core via fp32 WMMA (K = 95 padded to 96 -> 24 x K4 steps);
// i=16 row / j=16 col border + lvl1 via VALU on lanes 0..16.
// ---------------------------------------------------------------------------
__global__ __launch_bounds__(256) void k_path_sig(
    const float* __restrict__ x, const float* __restrict__ cw,
    const float* __restrict__ cb, float* __restrict__ S)
{
  __shared__ float cwsh[NK*NV*9];
  __shared__ float cbsh[NK];
  __shared__ float lp[8][NT*NF];   // per-wave path buffer p[t][i]

  const int tid  = threadIdx.x;
  const int wave = tid >> 5;
  const int lane = tid & 31;

  for (int q = tid; q < NK*NV*9; q += 256) cwsh[q] = cw[q];
  if (tid < NK) cbsh[tid] = cb[tid];
  __syncthreads();

  const int pix = blockIdx.x * 8 + wave;    // = b*HW + hw, grid exact
  const int b  = pix / HW;
  const int hw = pix % HW;
  const int h  = hw / W_, w = hw % W_;
  float* lpw = lp[wave];

  // Build path rows: lanes cover t = lane, lane+32, lane+64 (uniform 3 iters).
  for (int t = lane; t < NT; t += 32) {
    const float* xi = x + ((size_t)(b*NT + t) * NV) * HW;
    float acc[NK];
    #pragma unroll
    for (int k = 0; k < NK; ++k) acc[k] = cbsh[k];
    #pragma unroll
    for (int v = 0; v < NV; ++v) {
      #pragma unroll
      for (int dy = -1; dy <= 1; ++dy) {
        int hh = h + dy;
        if (hh < 0 || hh >= H_) continue;
        #pragma unroll
        for (int dxx = -1; dxx <= 1; ++dxx) {
          int ww = w + dxx;
          if (ww < 0 || ww >= W_) continue;
          float xv = xi[v*HW + hh*W_ + ww];
          const float* wp = &cwsh[v*9 + (dy+1)*3 + (dxx+1)];
          #pragma unroll
          for (int k = 0; k < NK; ++k) acc[k] = fmaf(xv, wp[k*36], acc[k]);
        }
      }
    }
    float* row = lpw + t*NF;
    #pragma unroll
    for (int k = 0; k < NK; ++k) row[k] = acc[k];
    #pragma unroll
    for (int v = 0; v < NV; ++v) row[NK+v] = xi[v*HW + h*W_ + w];
    row[NK+NV] = (float)t * (1.0f/95.0f);   // tch = linspace(0,1,96)
  }
  __syncthreads();

  // ---- lvl2[0:16,0:16] via V_WMMA_F32_16X16X4_F32 ----
  // A[i][t] = 0.5*(p[t+1]+p[t])[i] - p[0][i]; B[t][j] = (p[t+1]-p[t])[j].
  // Lane&15 serves as both M (A rows) and N (B cols); hi selects K pair.
  const int i  = lane & 15;
  const int hi = lane >> 4;
  const float p0 = lpw[i];
  v8f c = {0.f,0.f,0.f,0.f,0.f,0.f,0.f,0.f};
  for (int kk = 0; kk < 24; ++kk) {
    int t0 = 4*kk + 2*hi;
    int t2 = t0 + 2; if (t2 > 95) t2 = 95;   // pad t=95: dx=0 contribution
    float pa = lpw[t0*NF + i];
    float pb = lpw[(t0+1)*NF + i];
    float pc = lpw[t2*NF + i];
    v2f af, bf;
    af.x = 0.5f*(pb + pa) - p0;   af.y = 0.5f*(pc + pb) - p0;
    bf.x = pb - pa;               bf.y = pc - pb;
    c = __builtin_amdgcn_wmma_f32_16x16x4_f32(
        false, af, false, bf, (short)0, c, false, false);
  }

  float* sp = S + (size_t)pix * SIGLEN;
  #pragma unroll
  for (int v = 0; v < 8; ++v) {
    int row = v + 8*hi;                 // C/D layout: VGPR v -> rows v, v+8
    sp[NF + row*NF + i] = c[v];
  }

  // ---- border (row i=16, col j=16) + lvl1, lanes 0..16 ----
  if (lane < NF) {
    int j = lane;
    float p0j  = lpw[j];
    float p016 = lpw[16];
    float rowacc = 0.f, colacc = 0.f;
    for (int t = 0; t < 95; ++t) {
      float ptj  = lpw[t*NF + j],  pt1j  = lpw[(t+1)*NF + j];
      float pt6  = lpw[t*NF + 16], pt16  = lpw[(t+1)*NF + 16];
      rowacc = fmaf(0.5f*(pt16 + pt6) - p016, pt1j - ptj, rowacc);  // lvl2[16][j]
      colacc = fmaf(0.5f*(pt1j + ptj) - p0j,  pt16 - pt6, colacc);  // lvl2[j][16]
    }
    sp[NF + 16*NF + j] = rowacc;
    if (j < 16) sp[NF + j*NF + 16] = colacc;
    sp[j] = lpw[95*NF + j] - p0j;       // lvl1
  }
}

// ---------------------------------------------------------------------------
// Kernel 2: MLP layer-1 GEMM, split-K over hw.  flat k = hw*306 + f.
// One block per hw (400 blocks); 8 waves = 4 M-tiles (batch) x 2 N-tiles.
// Each wave: 77 x K4 fp32 WMMA over f = 0..305 (padded to 308).
// Deterministic partials to P1[hw][64][32].
// ---------------------------------------------------------------------------
__global__ __launch_bounds__(256) void k_gemm1(
    const float* __restrict__ S, const float* __restrict__ w1,
    float* __restrict__ P1)
{
  const int hw   = blockIdx.x;
  const int wave = threadIdx.x >> 5;
  const int lane = threadIdx.x & 31;
  const int mt = wave >> 1;           // batch tile 0..3
  const int nt = wave & 1;            // output tile 0..1
  const int i  = lane & 15;
  const int hi = lane >> 4;

  const float* Sb = S  + ((size_t)(mt*16 + i) * HW + hw) * SIGLEN;
  const float* Wb = w1 + (size_t)hw * SIGLEN * 32 + (nt*16 + i);

  v8f c = {0.f,0.f,0.f,0.f,0.f,0.f,0.f,0.f};
  for (int kk = 0; kk < 77; ++kk) {
    int k0 = 4*kk + 2*hi;
    int k1 = k0 + 1;
    v2f a, bf;
    a.x  = (k0 < SIGLEN) ? Sb[k0] : 0.f;
    a.y  = (k1 < SIGLEN) ? Sb[k1] : 0.f;
    bf.x = (k0 < SIGLEN) ? Wb[(size_t)k0 * 32] : 0.f;
    bf.y = (k1 < SIGLEN) ? Wb[(size_t)k1 * 32] : 0.f;
    c = __builtin_amdgcn_wmma_f32_16x16x4_f32(
        false, a, false, bf, (short)0, c, false, false);
  }

  float* dst = P1 + (size_t)hw * (NB*32);
  #pragma unroll
  for (int v = 0; v < 8; ++v)
    dst[(mt*16 + v + 8*hi)*32 + (nt*16 + i)] = c[v];
}

// ---------------------------------------------------------------------------
// Kernel 3: reduce split-K partials (fixed order -> deterministic), bias+ReLU,
// then tiny layers 2-4 (one thread per batch row).
// ---------------------------------------------------------------------------
__global__ __launch_bounds__(256) void k_mlp_tail(
    const float* __restrict__ P1,
    const float* __restrict__ b1, const float* __restrict__ w2,
    const float* __restrict__ b2, const float* __restrict__ w3,
    const float* __restrict__ b3, const float* __restrict__ w4,
    const float* __restrict__ b4, float* __restrict__ out)
{
  __shared__ float h1[NB*33];   // stride 33 to dodge bank conflicts
  const int tid = threadIdx.x;

  for (int idx = tid; idx < NB*32; idx += 256) {
    int b = idx >> 5, n = idx & 31;
    float s = b1[n];
    for (int hw = 0; hw < HW; ++hw)
      s += P1[((size_t)hw*NB + b)*32 + n];
    h1[b*33 + n] = fmaxf(s, 0.f);
  }
  __syncthreads();

  if (tid < NB) {
    const float* hr = &h1[tid*33];
    float h2[32], h3[32];
    #pragma unroll 4
    for (int n = 0; n < 32; ++n) {
      float s = b2[n];
      for (int m = 0; m < 32; ++m) s = fmaf(hr[m], w2[m*32 + n], s);
      h2[n] = fmaxf(s, 0.f);
    }
    #pragma unroll 4
    for (int n = 0; n < 32; ++n) {
      float s = b3[n];
      for (int m = 0; m < 32; ++m) s = fmaf(h2[m], w3[m*32 + n], s);
      h3[n] = fmaxf(s, 0.f);
    }
    float o = b4[0];
    for (int m = 0; m < 32; ++m) o = fmaf(h3[m], w4[m], o);
    out[tid] = o;
  }
}

// ---------------------------------------------------------------------------
extern "C" void kernel_launch(void* const* d_in, const int* in_sizes, int n_in,
                              void* d_out, int out_size, void* d_ws, size_t ws_size,
                              hipStream_t stream)
{
  const float* x  = (const float*)d_in[0];
  const float* cw = (const float*)d_in[1];
  const float* cb = (const float*)d_in[2];
  const float* w1 = (const float*)d_in[3];
  const float* b1 = (const float*)d_in[4];
  const float* w2 = (const float*)d_in[5];
  const float* b2 = (const float*)d_in[6];
  const float* w3 = (const float*)d_in[7];
  const float* b3 = (const float*)d_in[8];
  const float* w4 = (const float*)d_in[9];
  const float* b4 = (const float*)d_in[10];
  float* out = (float*)d_out;

  // Workspace: sig S [25600][306] (31.3 MB) + split-K partials P1 [400][64][32]
  float* S  = (float*)d_ws;
  float* P1 = S + (size_t)NPIX * SIGLEN;

  k_path_sig<<<NPIX/8, 256, 0, stream>>>(x, cw, cb, S);
  k_gemm1  <<<HW,      256, 0, stream>>>(S, w1, P1);
  k_mlp_tail<<<1,      256, 0, stream>>>(P1, b1, w2, b2, w3, b3, w4, b4, out);
}